// SymDir_69681549410630
// MI455X (gfx1250) — compile-verified
//
#include <hip/hip_runtime.h>
#include <hip/hip_bf16.h>

// SymDir loss reduction for MI455X (gfx1250, wave32).
// HBM-bound: ~185 MB streamed once, ~8 us floor at 23.3 TB/s.
// - Bulk loads are non-temporal (TH=NT): pure one-pass stream, no reuse.
// - Intra-wave reduction runs on V_WMMA_F32_16X16X4_F32 (full f32 precision;
//   ones-matrix B turns the MMA into a cross-lane sum). The WMMA leaves the
//   sum of lanes 0-7,16-23 in lanes 0-15 and that of lanes 8-15,24-31 in lanes 16-31, so lanes
//   0 and 16 write straight to LDS -- no ds_bpermute round trip needed.

typedef __attribute__((ext_vector_type(2))) float f2;
typedef __attribute__((ext_vector_type(4))) float f4;
typedef __attribute__((ext_vector_type(8))) float f8;

// Returns a value s such that lane 0 holds sum(acc[0..7]) + sum(acc[16..23]) and lane 16 holds
// sum(acc[8..15]) + sum(acc[24..31]) (WMMA path). Other lanes hold replicas of their half's sum.
__device__ __forceinline__ float wave_half_sums(float x) {
#if defined(__gfx1250__) && __has_builtin(__builtin_amdgcn_wmma_f32_16x16x4_f32)
    // A (16x4 f32): VGPR0 = x -> A[m][0] (lanes 0-15), A[m][2] (lanes 16-31);
    // VGPR1 = 0 -> A[m][1] = A[m][3] = 0. B = ones(4x16).
    // D[m][n] = acc_m + acc_{m+16} for every n.
    f2 av; av[0] = x;    av[1] = 0.0f;
    f2 bv; bv[0] = 1.0f; bv[1] = 1.0f;
    f8 cv = {0.0f, 0.0f, 0.0f, 0.0f, 0.0f, 0.0f, 0.0f, 0.0f};
    f8 dv = __builtin_amdgcn_wmma_f32_16x16x4_f32(
        /*neg_a=*/false, av, /*neg_b=*/false, bv,
        /*c_mod=*/(short)0, cv, /*reuse_a=*/false, /*reuse_b=*/false);
    // Lanes 0-15: VGPR r = D[r][n] (rows 0-7); lanes 16-31: rows 8-15.
    // Per-lane sum of the 8 D registers = half of the total wave sum.
    return ((dv[0] + dv[1]) + (dv[2] + dv[3])) + ((dv[4] + dv[5]) + (dv[6] + dv[7]));
#else
    // Reduce within each 16-lane half.
    for (int off = 8; off > 0; off >>= 1) x += __shfl_xor(x, off, 16);
    return x;
#endif
}

// Block of 256 threads = 8 wave32 waves -> 16 half-sums in LDS.
// Result valid in thread 0 only.
__device__ __forceinline__ float block_sum256(float x) {
    __shared__ float hsum[16];
    const float s = wave_half_sums(x);
    if ((threadIdx.x & 15) == 0)          // lanes 0 and 16 of each wave
        hsum[threadIdx.x >> 4] = s;
    __syncthreads();
    float tot = 0.0f;
    if (threadIdx.x == 0) {
        float t0 = ((hsum[0] + hsum[1]) + (hsum[2] + hsum[3])) +
                   ((hsum[4] + hsum[5]) + (hsum[6] + hsum[7]));
        float t1 = ((hsum[8] + hsum[9]) + (hsum[10] + hsum[11])) +
                   ((hsum[12] + hsum[13]) + (hsum[14] + hsum[15]));
        tot = t0 + t1;
    }
    return tot;
}

__global__ void __launch_bounds__(256)
symdir_partial_kernel(const float* __restrict__ y,     // [T,3,2]
                      const float* __restrict__ Ainv,  // [T,2,2]
                      const float* __restrict__ a,     // [T]
                      float* __restrict__ partial,     // [gridDim.x]
                      int T) {
    const int tid    = blockIdx.x * blockDim.x + threadIdx.x;
    const int stride = gridDim.x * blockDim.x;

    const f2* __restrict__ y2 = (const f2*)y;     // 8B-aligned per vertex
    const f4* __restrict__ A4 = (const f4*)Ainv;  // 16B-aligned per triangle

    float acc = 0.0f;
    for (int t = tid; t < T; t += stride) {
        // One-pass stream: non-temporal loads (global_load_b64/b128, TH=NT).
        const f2 p0 = __builtin_nontemporal_load(&y2[3 * t + 0]);
        const f2 p1 = __builtin_nontemporal_load(&y2[3 * t + 1]);
        const f2 p2 = __builtin_nontemporal_load(&y2[3 * t + 2]);
        const f4 Ai = __builtin_nontemporal_load(&A4[t]);   // (A00,A01,A10,A11)
        const float at = __builtin_nontemporal_load(&a[t]);

        // Edge matrix B (columns are edges)
        const float B00 = p1[0] - p0[0], B10 = p1[1] - p0[1];
        const float B01 = p2[0] - p0[0], B11 = p2[1] - p0[1];

        // J = B * Ainv
        const float J00 = B00 * Ai[0] + B01 * Ai[2];
        const float J01 = B00 * Ai[1] + B01 * Ai[3];
        const float J10 = B10 * Ai[0] + B11 * Ai[2];
        const float J11 = B10 * Ai[1] + B11 * Ai[3];

        const float fro2 = (J00 * J00 + J01 * J01) + (J10 * J10 + J11 * J11);
        const float det  = J00 * J11 - J01 * J10;
        const float d2   = det * det;

        acc += at * fro2 * (1.0f + 1.0f / d2);
    }

    const float bsum = block_sum256(acc);
    if (threadIdx.x == 0) partial[blockIdx.x] = bsum;
}

__global__ void __launch_bounds__(256)
symdir_final_kernel(const float* __restrict__ partial, int n,
                    float* __restrict__ out) {
    float acc = 0.0f;
    for (int i = threadIdx.x; i < n; i += blockDim.x) acc += partial[i];
    const float tot = block_sum256(acc);
    if (threadIdx.x == 0) out[0] = tot;  // plain write: deterministic per replay
}

extern "C" void kernel_launch(void* const* d_in, const int* in_sizes, int n_in,
                              void* d_out, int out_size, void* d_ws, size_t ws_size,
                              hipStream_t stream) {
    const float* y    = (const float*)d_in[0];  // [T,3,2] f32
    const float* Ainv = (const float*)d_in[1];  // [T,2,2] f32
    const float* a    = (const float*)d_in[2];  // [T]     f32
    const int    T    = in_sizes[2];            // element count of 'a'

    float* partial = (float*)d_ws;              // 2048 floats = 8 KB scratch

    const int NT = 256;                          // 8 wave32 waves per block
    const int NB = 2048;                         // 8 triangles/thread at T=4M

    symdir_partial_kernel<<<NB, NT, 0, stream>>>(y, Ainv, a, partial, T);
    symdir_final_kernel<<<1, NT, 0, stream>>>(partial, NB, (float*)d_out);
}